// DetectionLoss_66374424592475
// MI455X (gfx1250) — compile-verified
//
#include <hip/hip_runtime.h>
#include <hip/hip_bf16.h>
#include <math.h>

// ---------------- problem constants ----------------
#define BM   16      // boxes per image (M)
#define AA   3       // anchors per cell
#define CCLS 3       // classes
#define BB   32      // batch

typedef float v2f __attribute__((ext_vector_type(2)));
typedef float v8f __attribute__((ext_vector_type(8)));

// ======================================================================
// Kernel 0: zero the (contiguous) partials region so padded WMMA loads
// are unguarded and never see poisoned workspace.
// ======================================================================
__global__ __launch_bounds__(256)
void zero_parts(float* __restrict__ p, int n)
{
    int i = blockIdx.x * 256 + threadIdx.x;
    if (i < n) p[i] = 0.0f;
}

// ======================================================================
// Kernel 1: per-anchor pass (one launch per scale).
// grid = (ceil(HW/256), B, A); block = 256. Thread -> cell, blockIdx.z -> a.
// Emits: ordered uint key per anchor (neg -> bits|MSB, else low sentinel)
// and 5 per-block partial sums: {npos, navail, sum_obj_pos, sum_ce, sum_sl1}.
// ======================================================================
__global__ __launch_bounds__(256)
void anchor_kernel(const float* __restrict__ pred,   // (B, A*8, H, W)
                   const float* __restrict__ anc,    // (Ns, 4) cx,cy,s,s
                   const float* __restrict__ boxes,  // (B, 16, 4)
                   const int*   __restrict__ labels, // (B, 16)
                   unsigned*    __restrict__ keys,   // (B, Ns)
                   float*       __restrict__ part,   // (5, B, pad)
                   int HW, int Ns, int pad)
{
    __shared__ float sbx0[BM], sby0[BM], sbx1[BM], sby1[BM], sba[BM];
    __shared__ float sbcx[BM], sbcy[BM], sbw[BM], sbh[BM];
    __shared__ int   slab[BM];
    __shared__ float red[256];

    const int tid = threadIdx.x;
    const int b   = blockIdx.y;
    const int a   = blockIdx.z;

    if (tid < BM) {
        const float* bb = boxes + ((size_t)b * BM + tid) * 4;
        float cx = bb[0], cy = bb[1], bw = bb[2], bh = bb[3];
        float x0 = cx - bw * 0.5f, y0 = cy - bh * 0.5f;
        float x1 = cx + bw * 0.5f, y1 = cy + bh * 0.5f;
        sbx0[tid] = x0; sby0[tid] = y0; sbx1[tid] = x1; sby1[tid] = y1;
        sba[tid]  = (x1 - x0) * (y1 - y0);
        sbcx[tid] = cx; sbcy[tid] = cy; sbw[tid] = bw; sbh[tid] = bh;
        slab[tid] = labels[b * BM + tid];
    }
    __syncthreads();

    const int cell = blockIdx.x * 256 + tid;
    float v_npos = 0.f, v_navail = 0.f, v_obj = 0.f, v_ce = 0.f, v_sl1 = 0.f;

    if (cell < HW) {
        const int i = cell * AA + a;          // anchor index in (h,w,a) order
        const float* ap = anc + (size_t)i * 4;
        float acx = ap[0], acy = ap[1], asz = ap[2];
        float ax0 = acx - asz * 0.5f, ay0 = acy - asz * 0.5f;
        float ax1 = acx + asz * 0.5f, ay1 = acy + asz * 0.5f;
        float areaA = (ax1 - ax0) * (ay1 - ay0);

        float best = -1.0f; int midx = 0;
        #pragma unroll
        for (int m = 0; m < BM; ++m) {
            float ltx = fmaxf(ax0, sbx0[m]);
            float lty = fmaxf(ay0, sby0[m]);
            float rbx = fminf(ax1, sbx1[m]);
            float rby = fminf(ay1, sby1[m]);
            float iw  = fmaxf(rbx - ltx, 0.f);
            float ih  = fmaxf(rby - lty, 0.f);
            float inter = iw * ih;
            float iou = inter / (areaA + sba[m] - inter + 1e-9f);
            if (iou > best) { best = iou; midx = m; }   // first-max like argmax
        }
        bool pos = (best >= 0.5f);
        bool neg = (best <  0.4f);

        // pred channels for this anchor: base channel a*8, plane stride HW
        const float* pb = pred + ((size_t)b * (AA * 8) + (size_t)a * 8) * HW + cell;

        float o = pb[(size_t)4 * HW];
        float obj_all = fmaxf(o, 0.f) - (pos ? o : 0.f) + log1pf(expf(-fabsf(o)));

        unsigned kb = 0x007FFFFFu;                       // sentinel (below all neg keys)
        if (neg) kb = __float_as_uint(obj_all) | 0x80000000u;  // obj_all >= 0
        keys[(size_t)b * Ns + i] = kb;

        v_npos   = pos ? 1.f : 0.f;
        v_navail = neg ? 1.f : 0.f;
        v_obj    = pos ? obj_all : 0.f;

        if (pos) {
            // class CE (log-softmax over 3 logits)
            float c0 = pb[(size_t)5 * HW], c1 = pb[(size_t)6 * HW], c2 = pb[(size_t)7 * HW];
            float mx  = fmaxf(c0, fmaxf(c1, c2));
            float lse = mx + logf(expf(c0 - mx) + expf(c1 - mx) + expf(c2 - mx));
            int t = slab[midx] - 1;
            t = (t < 0) ? 0 : ((t > CCLS - 1) ? CCLS - 1 : t);
            float ct = (t == 0) ? c0 : ((t == 1) ? c1 : c2);
            v_ce = lse - ct;

            // SmoothL1 box regression
            float t0 = (sbcx[midx] - acx) / asz;
            float t1 = (sbcy[midx] - acy) / asz;
            float t2 = logf(sbw[midx] / asz);
            float t3 = logf(sbh[midx] / asz);
            float l0 = pb[0];
            float l1 = pb[(size_t)1 * HW];
            float l2 = pb[(size_t)2 * HW];
            float l3 = pb[(size_t)3 * HW];
            float d0 = fabsf(l0 - t0), d1 = fabsf(l1 - t1);
            float d2 = fabsf(l2 - t2), d3 = fabsf(l3 - t3);
            float s0 = (d0 < 1.f) ? 0.5f * d0 * d0 : d0 - 0.5f;
            float s1 = (d1 < 1.f) ? 0.5f * d1 * d1 : d1 - 0.5f;
            float s2 = (d2 < 1.f) ? 0.5f * d2 * d2 : d2 - 0.5f;
            float s3 = (d3 < 1.f) ? 0.5f * d3 * d3 : d3 - 0.5f;
            v_sl1 = s0 + s1 + s2 + s3;
        }
    }

    // deterministic block reductions -> partials
    float vals[5] = { v_npos, v_navail, v_obj, v_ce, v_sl1 };
    #pragma unroll
    for (int q = 0; q < 5; ++q) {
        red[tid] = vals[q];
        __syncthreads();
        for (int off = 128; off >= 1; off >>= 1) {
            if (tid < off) red[tid] += red[tid + off];
            __syncthreads();
        }
        if (tid == 0)
            part[((size_t)q * gridDim.y + b) * pad + (size_t)a * gridDim.x + blockIdx.x] = red[0];
        __syncthreads();
    }
}

// ======================================================================
// Kernel 2: per-segment finalize. grid = (B, 3). block = 256.
// Wave 0 reduces block partials with chained V_WMMA_F32_16X16X4_F32
// (A = ones => D row 0 = exact f32 column sums; slot assignment of B is
// sum-invariant, so each lane takes a contiguous unguarded b64 load).
// Then exact k-th largest neg BCE value via monotone bitwise search on
// ordered uint keys, then selected-negative sum/count, then losses.
// ======================================================================
__global__ __launch_bounds__(256)
void segment_kernel(const unsigned* __restrict__ keys0,
                    const unsigned* __restrict__ keys1,
                    const unsigned* __restrict__ keys2,
                    const float* __restrict__ part0,
                    const float* __restrict__ part1,
                    const float* __restrict__ part2,
                    float* __restrict__ seg_out)   // (3, B, 3)
{
    const int tid = threadIdx.x;
    const int b   = blockIdx.x;
    const int s   = blockIdx.y;

    const int NsT[3]   = { 76800, 19200, 4800 };
    const int padT[3]  = { 320,   128,   64   };   // zero-padded block counts

    const unsigned* keys = (s == 0) ? keys0 : (s == 1) ? keys1 : keys2;
    const float*    part = (s == 0) ? part0 : (s == 1) ? part1 : part2;
    const int Ns   = NsT[s];
    const int pad  = padT[s];
    keys += (size_t)b * Ns;

    __shared__ float d0lds[5][32];
    __shared__ float sSums[5];
    __shared__ int   sK;
    __shared__ int   redI[256];
    __shared__ float redF[256];

    // ---- WMMA f32 reduction of the 5 partial streams (wave 0 only) ----
    if (tid < 32) {
        v2f aones; aones[0] = 1.0f; aones[1] = 1.0f;
        for (int q = 0; q < 5; ++q) {
            const float* p = part + ((size_t)q * BB + b) * pad;
            v8f c = {};
            for (int base = 0; base < pad; base += 64) {
                v2f bv = *(const v2f*)(p + base + 2 * tid);   // contiguous b64 per lane
                c = __builtin_amdgcn_wmma_f32_16x16x4_f32(
                        false, aones, false, bv, (short)0, c, false, false);
            }
            d0lds[q][tid] = c[0];   // D row 0 (lanes 0..15) = column sums
        }
    }
    __syncthreads();
    if (tid == 0) {
        for (int q = 0; q < 5; ++q) {
            float t = 0.f;
            for (int l = 0; l < 16; ++l) t += d0lds[q][l];
            sSums[q] = t;
        }
        int npos  = (int)(sSums[0] + 0.5f);
        int avail = (int)(sSums[1] + 0.5f);
        int k = (npos == 0) ? ((avail < 100) ? avail : 100)
                            : ((avail < 3 * npos) ? avail : 3 * npos);
        sK = k;
    }
    __syncthreads();
    const int k = sK;

    // ---- exact k-th largest neg key: MSB-first monotone bit search ----
    unsigned cur = 0u;
    if (k > 0) {
        for (int bit = 31; bit >= 0; --bit) {
            unsigned cand = cur | (1u << bit);
            int local = 0;
            for (int idx = tid; idx < Ns; idx += 256)
                local += (keys[idx] >= cand) ? 1 : 0;
            redI[tid] = local;
            __syncthreads();
            for (int off = 128; off >= 1; off >>= 1) {
                if (tid < off) redI[tid] += redI[tid + off];
                __syncthreads();
            }
            int total = redI[0];
            __syncthreads();
            if (total >= k) cur = cand;   // uniform across block
        }
    }

    // ---- selected negatives: count + sum (thr with ties, like reference) ----
    float lsum = 0.f; int lcnt = 0;
    if (k > 0) {
        for (int idx = tid; idx < Ns; idx += 256) {
            unsigned kk = keys[idx];
            if (kk >= cur) {                       // sentinels < 0x80000000 <= cur
                lcnt += 1;
                lsum += __uint_as_float(kk & 0x7FFFFFFFu);
            }
        }
    }
    redI[tid] = lcnt; redF[tid] = lsum;
    __syncthreads();
    for (int off = 128; off >= 1; off >>= 1) {
        if (tid < off) { redI[tid] += redI[tid + off]; redF[tid] += redF[tid + off]; }
        __syncthreads();
    }

    if (tid == 0) {
        float npos_f = sSums[0];
        float cnt    = npos_f + (float)redI[0];
        float num    = sSums[2] + redF[0];
        float l_obj  = (cnt    > 0.f) ? num      / cnt            : 0.f;
        float l_cls  = (npos_f > 0.f) ? sSums[3] / npos_f         : 0.f;
        float l_loc  = (npos_f > 0.f) ? sSums[4] / (npos_f * 4.f) : 0.f;
        float* o = seg_out + ((size_t)s * BB + b) * 3;
        o[0] = l_obj; o[1] = l_cls; o[2] = l_loc;
    }
}

// ======================================================================
// Kernel 3: final scalar reduction -> [lo, lc, ll, total] / B
// ======================================================================
__global__ void final_kernel(const float* __restrict__ seg, float* __restrict__ out)
{
    if (threadIdx.x == 0) {
        float lo = 0.f, lc = 0.f, ll = 0.f;
        for (int i = 0; i < 3 * BB; ++i) {
            lo += seg[i * 3 + 0];
            lc += seg[i * 3 + 1];
            ll += seg[i * 3 + 2];
        }
        lo /= (float)BB; lc /= (float)BB; ll /= (float)BB;
        out[0] = lo; out[1] = lc; out[2] = ll; out[3] = lo + lc + ll;
    }
}

// ======================================================================
extern "C" void kernel_launch(void* const* d_in, const int* in_sizes, int n_in,
                              void* d_out, int out_size, void* d_ws, size_t ws_size,
                              hipStream_t stream)
{
    const float* pred0  = (const float*)d_in[0];
    const float* pred1  = (const float*)d_in[1];
    const float* pred2  = (const float*)d_in[2];
    const float* anc0   = (const float*)d_in[3];
    const float* anc1   = (const float*)d_in[4];
    const float* anc2   = (const float*)d_in[5];
    const float* boxes  = (const float*)d_in[6];
    const int*   labels = (const int*)d_in[7];
    float* out = (float*)d_out;

    const int Ns0 = 76800, Ns1 = 19200, Ns2 = 4800;
    const int HW0 = 25600, HW1 = 6400,  HW2 = 1600;
    const int cb0 = 100,   cb1 = 25,    cb2 = 7;      // cell-blocks per (b, a)
    const int pad0 = 320,  pad1 = 128,  pad2 = 64;    // padded block counts

    char* ws = (char*)d_ws;
    unsigned* keys0 = (unsigned*)ws;  ws += (size_t)BB * Ns0 * sizeof(unsigned);
    unsigned* keys1 = (unsigned*)ws;  ws += (size_t)BB * Ns1 * sizeof(unsigned);
    unsigned* keys2 = (unsigned*)ws;  ws += (size_t)BB * Ns2 * sizeof(unsigned);
    float* part0 = (float*)ws;        ws += (size_t)5 * BB * pad0 * sizeof(float);
    float* part1 = (float*)ws;        ws += (size_t)5 * BB * pad1 * sizeof(float);
    float* part2 = (float*)ws;        ws += (size_t)5 * BB * pad2 * sizeof(float);
    float* seg   = (float*)ws;        ws += (size_t)3 * BB * 3 * sizeof(float);

    dim3 blk(256);

    // partials are laid out contiguously: zero them (incl. padding) first
    const int npart = 5 * BB * (pad0 + pad1 + pad2);
    zero_parts<<<dim3((npart + 255) / 256), blk, 0, stream>>>(part0, npart);

    anchor_kernel<<<dim3(cb0, BB, AA), blk, 0, stream>>>(pred0, anc0, boxes, labels,
                                                         keys0, part0, HW0, Ns0, pad0);
    anchor_kernel<<<dim3(cb1, BB, AA), blk, 0, stream>>>(pred1, anc1, boxes, labels,
                                                         keys1, part1, HW1, Ns1, pad1);
    anchor_kernel<<<dim3(cb2, BB, AA), blk, 0, stream>>>(pred2, anc2, boxes, labels,
                                                         keys2, part2, HW2, Ns2, pad2);
    segment_kernel<<<dim3(BB, 3), blk, 0, stream>>>(keys0, keys1, keys2,
                                                    part0, part1, part2, seg);
    final_kernel<<<1, 32, 0, stream>>>(seg, out);

    (void)in_sizes; (void)n_in; (void)out_size; (void)ws_size;
}